// Quantum_81604378624624
// MI455X (gfx1250) — compile-verified
//
#include <hip/hip_runtime.h>
#include <hip/hip_bf16.h>

typedef float v2f __attribute__((ext_vector_type(2)));
typedef float v8f __attribute__((ext_vector_type(8)));

#define ISQ2 0.70710678118654752440f

// ---------------- workspace layout (in floats) ----------------
// conv raw buffer (max: L1 2048*3*1220 = 7,495,680)
static constexpr size_t OFF_CONV  = 0;
static constexpr size_t OFF_PING  = 7500000;   // max 3,747,840
static constexpr size_t OFF_PONG  = 11250000;  // max 3,102,720
static constexpr size_t OFF_QEXP  = 14360000;  // 2048*9 = 18,432
static constexpr size_t OFF_STATS = 14380000;  // 5 layers * 32 floats

// =====================================================================
// Conv + bias + ReLU + BN batch-stats (sum, sumsq), VALU path, layers 1-3.
// Each thread computes all OUTC channels for one (b,pos): input window reuse.
// =====================================================================
template<int INC, int OUTC, int KW, int STRIDE>
__global__ void conv_relu_stats_valu(const float* __restrict__ in,
                                     const float* __restrict__ w,
                                     const float* __restrict__ bias,
                                     float* __restrict__ out,
                                     float* __restrict__ stats,
                                     int B, int inL, int convL) {
  __shared__ float sw[INC * OUTC * KW];
  __shared__ float sb[OUTC];
  __shared__ float ssum[OUTC];
  __shared__ float ssq[OUTC];
  for (int i = threadIdx.x; i < INC * OUTC * KW; i += blockDim.x) sw[i] = w[i];
  if (threadIdx.x < OUTC) {
    sb[threadIdx.x]   = bias[threadIdx.x];
    ssum[threadIdx.x] = 0.f;
    ssq[threadIdx.x]  = 0.f;
  }
  __syncthreads();

  float lsum[OUTC], lsq[OUTC];
#pragma unroll
  for (int c = 0; c < OUTC; ++c) { lsum[c] = 0.f; lsq[c] = 0.f; }

  const int total = B * convL;
  for (int idx = blockIdx.x * blockDim.x + threadIdx.x; idx < total;
       idx += gridDim.x * blockDim.x) {
    const int b   = idx / convL;
    const int pos = idx - b * convL;
    const float* ip = in + b * INC * inL + pos * STRIDE;
    float xv[INC * KW];
#pragma unroll
    for (int i = 0; i < INC; ++i)
#pragma unroll
      for (int kh = 0; kh < KW; ++kh) xv[i * KW + kh] = ip[i * inL + kh];
#pragma unroll
    for (int c = 0; c < OUTC; ++c) {
      float s = sb[c];
#pragma unroll
      for (int t = 0; t < INC * KW; ++t) s += sw[c * INC * KW + t] * xv[t];
      s = fmaxf(s, 0.f);
      out[b * OUTC * convL + c * convL + pos] = s;
      lsum[c] += s;
      lsq[c]  += s * s;
    }
  }
#pragma unroll
  for (int c = 0; c < OUTC; ++c) {
    atomicAdd(&ssum[c], lsum[c]);
    atomicAdd(&ssq[c],  lsq[c]);
  }
  __syncthreads();
  if (threadIdx.x < OUTC) {
    atomicAdd(&stats[threadIdx.x],      ssum[threadIdx.x]);
    atomicAdd(&stats[16 + threadIdx.x], ssq[threadIdx.x]);
  }
}

// =====================================================================
// Conv + bias + ReLU + BN batch-stats via f32 WMMA, layers 4-5 (outC==16).
// GEMM: D(16x16) = W(16xK) x Im2col(Kx16), one wave per (16-batch, pos) tile.
// A layout (16x4 f32): lanes 0-15 hold K={kk,kk+1} in v0/v1; lanes 16-31 K={kk+2,kk+3}.
// B layout (4x16 f32): lane = N, same K-half split across lane halves.
// D layout: VGPR r -> M=r (lanes 0-15) / M=r+8 (lanes 16-31), N = lane&15.
// =====================================================================
template<int INC, int KW, int INL, int CONVL>
__global__ void conv_relu_stats_wmma(const float* __restrict__ in,
                                     const float* __restrict__ w,
                                     const float* __restrict__ bias,
                                     float* __restrict__ out,
                                     float* __restrict__ stats,
                                     int B) {
  constexpr int K = INC * KW;  // 32 (L4) / 80 (L5), both % 4 == 0
  __shared__ float sw[16 * K];
  __shared__ float ssum[16];
  __shared__ float ssq[16];
  for (int i = threadIdx.x; i < 16 * K; i += blockDim.x) sw[i] = w[i];
  if (threadIdx.x < 16) { ssum[threadIdx.x] = 0.f; ssq[threadIdx.x] = 0.f; }
  __syncthreads();

  const int lane = threadIdx.x & 31;
  const int wave = threadIdx.x >> 5;
  const int half = lane >> 4;    // K-half selector
  const int nm   = lane & 15;    // = M row for A, = N col for B/D

  float lsum[8], lsq[8];
#pragma unroll
  for (int r = 0; r < 8; ++r) { lsum[r] = 0.f; lsq[r] = 0.f; }

  const int nTiles      = (B >> 4) * CONVL;
  const int wavesPerGrid = gridDim.x * (blockDim.x >> 5);

  for (int tile = blockIdx.x * (blockDim.x >> 5) + wave; tile < nTiles;
       tile += wavesPerGrid) {
    const int tb  = tile / CONVL;
    const int pos = tile - tb * CONVL;
    const int b0  = tb << 4;
    const float* ip = in + (b0 + nm) * INC * INL + pos;  // column n = nm
    const float* ap = sw + nm * K;                        // row m = nm

    v8f acc = {};
#pragma unroll
    for (int kk = 0; kk < K; kk += 4) {
      const int k0 = kk + 2 * half;
      v2f a;
      a.x = ap[k0];
      a.y = ap[k0 + 1];
      const int i0 = k0 / KW,       h0 = k0 - i0 * KW;
      const int i1 = (k0 + 1) / KW, h1 = (k0 + 1) - i1 * KW;
      v2f bb;
      bb.x = ip[i0 * INL + h0];
      bb.y = ip[i1 * INL + h1];
      acc = __builtin_amdgcn_wmma_f32_16x16x4_f32(
          false, a, false, bb, (short)0, acc, false, false);
    }
#pragma unroll
    for (int r = 0; r < 8; ++r) {
      const int ch = half * 8 + r;
      float v = acc[r] + bias[ch];
      v = fmaxf(v, 0.f);
      out[(b0 + nm) * 16 * CONVL + ch * CONVL + pos] = v;
      lsum[r] += v;
      lsq[r]  += v * v;
    }
  }
#pragma unroll
  for (int r = 0; r < 8; ++r) {
    const int ch = half * 8 + r;
    atomicAdd(&ssum[ch], lsum[r]);
    atomicAdd(&ssq[ch],  lsq[r]);
  }
  __syncthreads();
  if (threadIdx.x < 16) {
    atomicAdd(&stats[threadIdx.x],      ssum[threadIdx.x]);
    atomicAdd(&stats[16 + threadIdx.x], ssq[threadIdx.x]);
  }
}

// =====================================================================
// BN apply (batch stats, biased var, eps=1e-5) + optional MaxPool(2,1).
// =====================================================================
__global__ void bn_pool(const float* __restrict__ y,
                        const float* __restrict__ stats,
                        const float* __restrict__ g,
                        const float* __restrict__ beta,
                        float* __restrict__ out,
                        int B, int C, int convL, int pool) {
  __shared__ float sscale[16];
  __shared__ float sshift[16];
  if (threadIdx.x < C) {
    const float cnt  = (float)B * (float)convL;
    const float mu   = stats[threadIdx.x] / cnt;
    const float var  = stats[16 + threadIdx.x] / cnt - mu * mu;
    const float inv  = rsqrtf(var + 1e-5f);
    const float sc   = g[threadIdx.x] * inv;
    sscale[threadIdx.x] = sc;
    sshift[threadIdx.x] = beta[threadIdx.x] - mu * sc;
  }
  __syncthreads();
  const int outL  = pool ? (convL >> 1) : convL;
  const int total = B * C * outL;
  for (int idx = blockIdx.x * blockDim.x + threadIdx.x; idx < total;
       idx += gridDim.x * blockDim.x) {
    const int i = idx % outL;
    const int c = (idx / outL) % C;
    const int b = idx / (outL * C);
    const float* yp = y + b * C * convL + c * convL;
    float v;
    if (pool) {
      const float v0 = sscale[c] * yp[2 * i]     + sshift[c];
      const float v1 = sscale[c] * yp[2 * i + 1] + sshift[c];
      v = fmaxf(v0, v1);
    } else {
      v = sscale[c] * yp[i] + sshift[c];
    }
    out[idx] = v;  // == b*C*outL + c*outL + i
  }
}

// =====================================================================
// 9-qubit statevector simulator: 1 block per batch element, state in LDS.
// 5 blocks of: H on all 9 qubits, CRY(i->i+1) chain, RX on all 9 qubits.
// Qubit q lives at bit (8-q) of the flat 512-index.
// =====================================================================
__global__ void qml_kernel(const float* __restrict__ feat,
                           const float* __restrict__ qw,
                           float* __restrict__ qexp) {
  __shared__ float sre[512];
  __shared__ float sim[512];
  __shared__ float2 scs[85];  // per block j: 8 CRY (cos,sin) then 9 RX (cos,sin)
  __shared__ float sexp[9];

  const int b = blockIdx.x;
  const int t = threadIdx.x;  // 256 threads

  sre[t]       = feat[b * 512 + t];
  sre[t + 256] = feat[b * 512 + t + 256];
  sim[t]       = 0.f;
  sim[t + 256] = 0.f;
  if (t < 85) {
    const int j = t / 17, gidx = t - j * 17;
    float th;
    if (gidx < 8) th = 0.5f * qw[j * 18 + gidx * 2 + 1];        // CRY theta
    else          th = 0.5f * qw[j * 18 + (gidx - 8) * 2 + 0];  // RX theta
    scs[t] = make_float2(cosf(th), sinf(th));
  }
  if (t < 9) sexp[t] = 0.f;

  for (int j = 0; j < 5; ++j) {
    // --- Hadamard on every qubit ---
#pragma unroll
    for (int q = 0; q < 9; ++q) {
      const int bp = 8 - q;
      const int m  = 1 << bp;
      __syncthreads();
      const int i0 = ((t >> bp) << (bp + 1)) | (t & (m - 1));
      const int i1 = i0 | m;
      const float r0 = sre[i0], r1 = sre[i1];
      const float m0 = sim[i0], m1 = sim[i1];
      sre[i0] = ISQ2 * (r0 + r1);
      sre[i1] = ISQ2 * (r0 - r1);
      sim[i0] = ISQ2 * (m0 + m1);
      sim[i1] = ISQ2 * (m0 - m1);
    }
    // --- CRY chain: control qubit i (bit bt+1), target qubit i+1 (bit bt) ---
#pragma unroll
    for (int i = 0; i < 8; ++i) {
      const int bt = 7 - i;
      const int m  = 1 << bt;
      const float2 cs = scs[j * 17 + i];
      __syncthreads();
      if (t < 128) {
        const int i0 = ((t >> bt) << (bt + 2)) | (1 << (bt + 1)) | (t & (m - 1));
        const int i1 = i0 | m;
        const float r0 = sre[i0], r1 = sre[i1];
        const float m0 = sim[i0], m1 = sim[i1];
        sre[i0] = cs.x * r0 - cs.y * r1;
        sre[i1] = cs.y * r0 + cs.x * r1;
        sim[i0] = cs.x * m0 - cs.y * m1;
        sim[i1] = cs.y * m0 + cs.x * m1;
      }
    }
    // --- RX on every qubit ---
#pragma unroll
    for (int i = 0; i < 9; ++i) {
      const int bp = 8 - i;
      const int m  = 1 << bp;
      const float2 cs = scs[j * 17 + 8 + i];
      __syncthreads();
      const int i0 = ((t >> bp) << (bp + 1)) | (t & (m - 1));
      const int i1 = i0 | m;
      const float r0 = sre[i0], r1 = sre[i1];
      const float m0 = sim[i0], m1 = sim[i1];
      sre[i0] = cs.x * r0 + cs.y * m1;
      sim[i0] = cs.x * m0 - cs.y * r1;
      sre[i1] = cs.x * r1 + cs.y * m0;
      sim[i1] = cs.x * m1 - cs.y * r0;
    }
  }
  __syncthreads();
  const float p0 = sre[t] * sre[t] + sim[t] * sim[t];
  const float p1 = sre[t + 256] * sre[t + 256] + sim[t + 256] * sim[t + 256];
#pragma unroll
  for (int i = 0; i < 9; ++i) {
    const int bp = 8 - i;
    float c;
    if (bp == 8) {
      c = p0 - p1;  // bit 8 of t is 0, of t+256 is 1
    } else {
      const float s = p0 + p1;
      c = ((t >> bp) & 1) ? -s : s;
    }
    atomicAdd(&sexp[i], c);
  }
  __syncthreads();
  if (t < 9) qexp[b * 9 + t] = sexp[t];
}

// =====================================================================
// Final FC: (B,9) @ (2,9)^T + bias
// =====================================================================
__global__ void fc_kernel(const float* __restrict__ qexp,
                          const float* __restrict__ fw,
                          const float* __restrict__ fb,
                          float* __restrict__ out, int B) {
  const int b = blockIdx.x * blockDim.x + threadIdx.x;
  if (b >= B) return;
  float a0 = fb[0], a1 = fb[1];
#pragma unroll
  for (int i = 0; i < 9; ++i) {
    const float q = qexp[b * 9 + i];
    a0 += q * fw[i];
    a1 += q * fw[9 + i];
  }
  out[b * 2 + 0] = a0;
  out[b * 2 + 1] = a1;
}

// =====================================================================
extern "C" void kernel_launch(void* const* d_in, const int* in_sizes, int n_in,
                              void* d_out, int out_size, void* d_ws, size_t ws_size,
                              hipStream_t stream) {
  (void)in_sizes; (void)n_in; (void)out_size; (void)ws_size;
  const float* x   = (const float*)d_in[0];
  const float* c1w = (const float*)d_in[1];
  const float* c1b = (const float*)d_in[2];
  const float* g1  = (const float*)d_in[3];
  const float* b1  = (const float*)d_in[4];
  const float* c2w = (const float*)d_in[5];
  const float* c2b = (const float*)d_in[6];
  const float* g2  = (const float*)d_in[7];
  const float* b2  = (const float*)d_in[8];
  const float* c3w = (const float*)d_in[9];
  const float* c3b = (const float*)d_in[10];
  const float* g3  = (const float*)d_in[11];
  const float* b3  = (const float*)d_in[12];
  const float* c4w = (const float*)d_in[13];
  const float* c4b = (const float*)d_in[14];
  const float* g4  = (const float*)d_in[15];
  const float* b4  = (const float*)d_in[16];
  const float* c5w = (const float*)d_in[17];
  const float* c5b = (const float*)d_in[18];
  const float* g5  = (const float*)d_in[19];
  const float* b5  = (const float*)d_in[20];
  const float* qwt = (const float*)d_in[21];
  const float* fcw = (const float*)d_in[22];
  const float* fcb = (const float*)d_in[23];

  float* ws    = (float*)d_ws;
  float* conv  = ws + OFF_CONV;
  float* ping  = ws + OFF_PING;
  float* pong  = ws + OFF_PONG;
  float* qe    = ws + OFF_QEXP;
  float* stats = ws + OFF_STATS;

  const int B = 2048;
  hipMemsetAsync(stats, 0, 5 * 32 * sizeof(float), stream);

  // L1: 1ch@1224 -> 3ch@1220 -> pool 610
  conv_relu_stats_valu<1, 3, 5, 1><<<1024, 256, 0, stream>>>(
      x, c1w, c1b, conv, stats + 0, B, 1224, 1220);
  bn_pool<<<1024, 256, 0, stream>>>(conv, stats + 0, g1, b1, ping, B, 3, 1220, 1);

  // L2: 3ch@610 -> 5ch@606 -> pool 303
  conv_relu_stats_valu<3, 5, 5, 1><<<1024, 256, 0, stream>>>(
      ping, c2w, c2b, conv, stats + 32, B, 610, 606);
  bn_pool<<<1024, 256, 0, stream>>>(conv, stats + 32, g2, b2, pong, B, 5, 606, 1);

  // L3: 5ch@303 -> 8ch@150 (stride 2) -> pool 75
  conv_relu_stats_valu<5, 8, 5, 2><<<1024, 256, 0, stream>>>(
      pong, c3w, c3b, conv, stats + 64, B, 303, 150);
  bn_pool<<<512, 256, 0, stream>>>(conv, stats + 64, g3, b3, ping, B, 8, 150, 1);

  // L4 (WMMA): 8ch@75 -> 16ch@72 -> pool 36   (tiles = 128*72 = 9216 waves)
  conv_relu_stats_wmma<8, 4, 75, 72><<<1152, 256, 0, stream>>>(
      ping, c4w, c4b, conv, stats + 96, B);
  bn_pool<<<512, 256, 0, stream>>>(conv, stats + 96, g4, b4, pong, B, 16, 72, 1);

  // L5 (WMMA): 16ch@36 -> 16ch@32 (no pool)   (tiles = 128*32 = 4096 waves)
  conv_relu_stats_wmma<16, 5, 36, 32><<<512, 256, 0, stream>>>(
      pong, c5w, c5b, conv, stats + 128, B);
  bn_pool<<<512, 256, 0, stream>>>(conv, stats + 128, g5, b5, ping, B, 16, 32, 0);

  // QML: 2048 statevectors of 512 amplitudes in LDS
  qml_kernel<<<2048, 256, 0, stream>>>(ping, qwt, qe);

  // FC 9 -> 2
  fc_kernel<<<8, 256, 0, stream>>>(qe, fcw, fcb, (float*)d_out, B);
}